// SAGEBATCH_33973191311569
// MI455X (gfx1250) — compile-verified
//
#include <hip/hip_runtime.h>
#include <hip/hip_bf16.h>

// ---------------------------------------------------------------------------
// GraphSAGE (3 layers) for MI455X / gfx1250.
//   layer: out = (mean-agg of neighbors) @ Wl^T + x @ Wr^T + b
// Phase 1 (per layer): edge scatter-add, wave-per-edge, lane-per-float4,
//                      f32 global atomics (agg + x rows live in 192MB L2).
// Phase 2 (per layer): fused dual-GEMM via V_WMMA_F32_16X16X4_F32 (full fp32),
//                      degree normalization folded into the A-operand load.
// ---------------------------------------------------------------------------

typedef float v2f __attribute__((ext_vector_type(2)));
typedef float v8f __attribute__((ext_vector_type(8)));

#define SAGE_N 100000
#define SAGE_E 3200000
#define SAGE_F 128

// ---- degree = segment_sum(ones, dst) --------------------------------------
__global__ void k_degree(const int* __restrict__ dst, float* __restrict__ deg,
                         long long nE) {
    long long e = (long long)blockIdx.x * blockDim.x + threadIdx.x;
    if (e < nE) atomicAdd(&deg[dst[e]], 1.0f);
}

// ---- deg -> 1/max(deg,1) ---------------------------------------------------
__global__ void k_invdeg(float* __restrict__ deg, int n) {
    int i = blockIdx.x * blockDim.x + threadIdx.x;
    if (i < n) deg[i] = 1.0f / fmaxf(deg[i], 1.0f);
}

// ---- edge scatter: agg[dst[e]] += h[src[e]]  (wave per edge) ---------------
// lane L of the wave moves bytes [16L, 16L+16) of the 512-byte feature row.
__global__ void k_scatter(const float* __restrict__ h,
                          const int* __restrict__ src,
                          const int* __restrict__ dst,
                          float* __restrict__ agg, long long nE) {
    long long idx = (long long)blockIdx.x * blockDim.x + threadIdx.x;
    long long e = idx >> 5;
    if (e >= nE) return;
    int lane = (int)(idx & 31);
    int s = src[e];
    int d = dst[e];
    const float4 v = *((const float4*)(h + (size_t)s * SAGE_F) + lane);
    float* o = agg + (size_t)d * SAGE_F + lane * 4;
    atomicAdd(o + 0, v.x);
    atomicAdd(o + 1, v.y);
    atomicAdd(o + 2, v.z);
    atomicAdd(o + 3, v.w);
}

// ---- fused SAGE GEMM: out = (agg .* invdeg) @ Wl^T + act @ Wr^T + bias -----
// One wave computes one 16x16 output tile with V_WMMA_F32_16X16X4_F32.
// mode 0: out0 = pre           (logits)
// mode 1: out0 = relu(pre)     (hidden layer 1)
// mode 2: out0 = pre (feature), out1 = relu(pre) (hidden layer 2)
__global__ void k_sage_gemm(const float* __restrict__ agg,
                            const float* __restrict__ invdeg,
                            const float* __restrict__ act,
                            const float* __restrict__ Wl,
                            const float* __restrict__ Wr,
                            const float* __restrict__ bias,
                            float* __restrict__ out0,
                            float* __restrict__ out1,
                            int outF, int mode, int nRowTiles) {
    const int K = SAGE_F;  // inner dim is always 128
    int lane = threadIdx.x & 31;
    int gid = (int)(((long long)blockIdx.x * blockDim.x + threadIdx.x) >> 5);
    int colTiles = outF >> 4;
    int tiles = nRowTiles * colTiles;
    if (gid >= tiles) return;  // wave-uniform: EXEC stays all-ones for WMMA

    int rowTile = gid / colTiles;
    int colTile = gid - rowTile * colTiles;
    int m0 = rowTile << 4;
    int n0 = colTile << 4;
    int half = lane >> 4;   // 0: K pair {0,1}, 1: K pair {2,3}
    int lid  = lane & 15;

    int rowA = m0 + lid;                 // A-matrix row owned by this lane
    float invd = invdeg[rowA];
    const float* aggRow = agg + (size_t)rowA * K;
    const float* actRow = act + (size_t)rowA * K;
    const float* wlRow  = Wl + (size_t)(n0 + lid) * K;  // B col = W row (W^T)
    const float* wrRow  = Wr + (size_t)(n0 + lid) * K;

    v8f c = {0.f, 0.f, 0.f, 0.f, 0.f, 0.f, 0.f, 0.f};

    // Path 1: mean-aggregated neighbors x Wl^T (normalization fused in)
    for (int k0 = 0; k0 < K; k0 += 4) {
        int ka = k0 + 2 * half;
        float2 av = *(const float2*)(aggRow + ka);
        float2 bv = *(const float2*)(wlRow + ka);
        v2f a = {av.x * invd, av.y * invd};
        v2f b = {bv.x, bv.y};
        c = __builtin_amdgcn_wmma_f32_16x16x4_f32(false, a, false, b,
                                                  (short)0, c, false, false);
    }
    // Path 2: self features x Wr^T
    for (int k0 = 0; k0 < K; k0 += 4) {
        int ka = k0 + 2 * half;
        float2 av = *(const float2*)(actRow + ka);
        float2 bv = *(const float2*)(wrRow + ka);
        v2f a = {av.x, av.y};
        v2f b = {bv.x, bv.y};
        c = __builtin_amdgcn_wmma_f32_16x16x4_f32(false, a, false, b,
                                                  (short)0, c, false, false);
    }

    int col = n0 + lid;
    float bval = bias[col];
#pragma unroll
    for (int j = 0; j < 8; ++j) {
        int row = m0 + j + 8 * half;     // C/D layout: VGPR j -> M = j + 8*half
        size_t o = (size_t)row * outF + col;
        float val = c[j] + bval;
        if (mode == 1) {
            out0[o] = fmaxf(val, 0.f);
        } else if (mode == 2) {
            out0[o] = val;               // pre-relu "feature"
            out1[o] = fmaxf(val, 0.f);   // feeds layer 3
        } else {
            out0[o] = val;               // logits
        }
    }
}

extern "C" void kernel_launch(void* const* d_in, const int* in_sizes, int n_in,
                              void* d_out, int out_size, void* d_ws, size_t ws_size,
                              hipStream_t stream) {
    const long long N = SAGE_N, E = SAGE_E;

    const float* x   = (const float*)d_in[0];
    const int*   ei  = (const int*)d_in[1];
    const int*   src = ei;
    const int*   dst = ei + E;
    const float* Wl1 = (const float*)d_in[2];
    const float* Wr1 = (const float*)d_in[3];
    const float* b1  = (const float*)d_in[4];
    const float* Wl2 = (const float*)d_in[5];
    const float* Wr2 = (const float*)d_in[6];
    const float* b2  = (const float*)d_in[7];
    const float* Wl3 = (const float*)d_in[8];
    const float* Wr3 = (const float*)d_in[9];
    const float* b3  = (const float*)d_in[10];

    // d_out: logits [N,64] then feature [N,128], flat in return order
    float* logits  = (float*)d_out;
    float* feature = (float*)d_out + N * 64;

    // workspace partition
    float* deg = (float*)d_ws;           // N       (becomes 1/max(deg,1))
    float* agg = deg + N;                // N*128   (reused each layer)
    float* h1  = agg + N * SAGE_F;       // N*128   relu(layer1)
    float* h2  = h1 + N * SAGE_F;        // N*128   relu(layer2)
    (void)in_sizes; (void)n_in; (void)out_size; (void)ws_size;

    const int TPB = 256;
    const int rowTiles = (int)(N / 16);  // 100000 = 6250*16 exactly

    // ---- degree (computed once, shared by all 3 layers) ----
    hipMemsetAsync(deg, 0, N * sizeof(float), stream);
    k_degree<<<(int)((E + TPB - 1) / TPB), TPB, 0, stream>>>(dst, deg, E);
    k_invdeg<<<(int)((N + TPB - 1) / TPB), TPB, 0, stream>>>(deg, (int)N);

    long long scatThreads = E * 32;
    int scatBlocks = (int)((scatThreads + TPB - 1) / TPB);

    auto gemmBlocks = [&](int outF) {
        long long waves = (long long)rowTiles * (outF / 16);
        return (int)((waves * 32 + TPB - 1) / TPB);
    };

    // ---- layer 1: x -> h1 (relu) ----
    hipMemsetAsync(agg, 0, N * SAGE_F * sizeof(float), stream);
    k_scatter<<<scatBlocks, TPB, 0, stream>>>(x, src, dst, agg, E);
    k_sage_gemm<<<gemmBlocks(128), TPB, 0, stream>>>(
        agg, deg, x, Wl1, Wr1, b1, h1, nullptr, 128, /*mode=*/1, rowTiles);

    // ---- layer 2: h1 -> feature (pre-relu, output #2) and h2 = relu ----
    hipMemsetAsync(agg, 0, N * SAGE_F * sizeof(float), stream);
    k_scatter<<<scatBlocks, TPB, 0, stream>>>(h1, src, dst, agg, E);
    k_sage_gemm<<<gemmBlocks(128), TPB, 0, stream>>>(
        agg, deg, h1, Wl2, Wr2, b2, feature, h2, 128, /*mode=*/2, rowTiles);

    // ---- layer 3: h2 -> logits (no relu) ----
    hipMemsetAsync(agg, 0, N * SAGE_F * sizeof(float), stream);
    k_scatter<<<scatBlocks, TPB, 0, stream>>>(h2, src, dst, agg, E);
    k_sage_gemm<<<gemmBlocks(64), TPB, 0, stream>>>(
        agg, deg, h2, Wl3, Wr3, b3, logits, nullptr, 64, /*mode=*/0, rowTiles);
}